// EncodingGraphTransformer_76630806495963
// MI455X (gfx1250) — compile-verified
//
#include <hip/hip_runtime.h>
#include <hip/hip_bf16.h>

#define N_NODES  100000
#define N_EDGES  800000
#define HIDN     64
#define NGROUPS  128
#define ENC_NEG_INF 0x007FFFFFu
#define NODE_TILES (N_NODES / 16)   // 6250 exactly

typedef float v2f __attribute__((ext_vector_type(2)));
typedef float v8f __attribute__((ext_vector_type(8)));
typedef __attribute__((address_space(3))) float lds_f32;

__device__ __forceinline__ unsigned lds_off(const float* p) {
  // ptrtoint of an addrspace(3) pointer == byte offset within the LDS alloc
  return (unsigned)(unsigned long long)(lds_f32*)(p);
}

__device__ __forceinline__ v8f wmma4(v2f a, v2f b, v8f c) {
  // V_WMMA_F32_16X16X4_F32 : D = A(16x4) x B(4x16) + C(16x16), fp32
  return __builtin_amdgcn_wmma_f32_16x16x4_f32(false, a, false, b, (short)0, c,
                                               false, false);
}

__device__ __forceinline__ float gelu_f(float x) {
  return 0.5f * x * (1.0f + erff(x * 0.7071067811865476f));
}

// order-preserving float<->uint for atomicMax on signed floats
__device__ __forceinline__ unsigned encf(float x) {
  unsigned u = __float_as_uint(x);
  return (u & 0x80000000u) ? ~u : (u | 0x80000000u);
}
__device__ __forceinline__ float decf(unsigned u) {
  return (u & 0x80000000u) ? __uint_as_float(u & 0x7FFFFFFFu)
                           : __uint_as_float(~u);
}

// ---------------------------------------------------------------------------
// Input projection: h = LN(gelu(x @ in_W + in_b)); x:[N,32], in_W:[32,64]
// One wave per 16-node tile; WMMA f32 16x16x4; LN via per-wave LDS tile.
// ---------------------------------------------------------------------------
__global__ void __launch_bounds__(256) k_inproj(
    const float* __restrict__ x, const float* __restrict__ W,
    const float* __restrict__ b, const float* __restrict__ lng,
    const float* __restrict__ lnb, float* __restrict__ hOut, int nTiles) {
  __shared__ float tile[8][16][65];
  int wslot = threadIdx.x >> 5;
  int wave  = blockIdx.x * 8 + wslot;
  if (wave >= nTiles) return;                 // wave-uniform guard
  int lane = threadIdx.x & 31;
  int row  = wave * 16 + (lane & 15);
  int kp   = (lane >> 4) * 2;
  int col  = lane & 15;

  v2f A[8];
  const float* xrow = x + row * 32;
#pragma unroll
  for (int ks = 0; ks < 8; ++ks) {
    A[ks].x = xrow[ks * 4 + kp];
    A[ks].y = xrow[ks * 4 + kp + 1];
  }
#pragma unroll
  for (int ct = 0; ct < 4; ++ct) {
    int c0 = ct * 16;
    float bv = b[c0 + col];
    v8f acc = {bv, bv, bv, bv, bv, bv, bv, bv};
#pragma unroll
    for (int ks = 0; ks < 8; ++ks) {
      v2f B;
      B.x = W[(ks * 4 + kp) * 64 + c0 + col];
      B.y = W[(ks * 4 + kp + 1) * 64 + c0 + col];
      acc = wmma4(A[ks], B, acc);
    }
#pragma unroll
    for (int r = 0; r < 8; ++r) {
      int mr = (lane >> 4) * 8 + r;
      tile[wslot][mr][c0 + col] = gelu_f(acc[r]);
    }
  }
  asm volatile("s_wait_dscnt 0x0" ::: "memory");  // wave-internal LDS fence
  if (lane < 16) {
    float mu = 0.f;
#pragma unroll
    for (int c = 0; c < 64; ++c) mu += tile[wslot][lane][c];
    mu *= (1.0f / 64.0f);
    float var = 0.f;
#pragma unroll
    for (int c = 0; c < 64; ++c) {
      float d = tile[wslot][lane][c] - mu;
      var += d * d;
    }
    var *= (1.0f / 64.0f);
    float inv = rsqrtf(var + 1e-5f);
    int rr = wave * 16 + lane;
#pragma unroll
    for (int c = 0; c < 64; ++c)
      hOut[rr * 64 + c] = (tile[wslot][lane][c] - mu) * inv * lng[c] + lnb[c];
  }
}

// ---------------------------------------------------------------------------
// 4 fused GEMMs: o_m = h @ W_m + b_m  (m = q,k,v,skip), all [N,64]x[64,64].
// Weights (64 KB) staged into LDS once per workgroup via async LDS loads
// (GLOBAL_LOAD_ASYNC_TO_LDS_B128, ASYNCcnt), then WMMA B fragments read from
// LDS. A fragments loaded once per wave, reused for 16 output tiles.
// ---------------------------------------------------------------------------
__global__ void __launch_bounds__(256) k_gemm4(
    const float* __restrict__ h, const float* __restrict__ Wqkv,  // [3*64*64]
    const float* __restrict__ Wskip,                              // [64*64]
    const float* __restrict__ b0, const float* __restrict__ b1,
    const float* __restrict__ b2, const float* __restrict__ b3,
    float* __restrict__ o0, float* __restrict__ o1,
    float* __restrict__ o2, float* __restrict__ o3, int nTiles) {
  __shared__ float smW[16384];  // [4][64][64]: q,k,v,skip
  {
    unsigned t = threadIdx.x;
    unsigned base = lds_off(&smW[0]);
    // q,k,v: 49152 B = 3072 x 16B chunks = 12 per thread
#pragma unroll
    for (int i = 0; i < 12; ++i) {
      unsigned chunk = t + i * 256u;
      unsigned dOff = base + chunk * 16u;
      unsigned gOff = chunk * 16u;
      asm volatile("global_load_async_to_lds_b128 %0, %1, %2"
                   :: "v"(dOff), "v"(gOff), "s"(Wqkv) : "memory");
    }
    // skip: 16384 B = 1024 x 16B chunks = 4 per thread
#pragma unroll
    for (int i = 0; i < 4; ++i) {
      unsigned chunk = t + i * 256u;
      unsigned dOff = base + 49152u + chunk * 16u;
      unsigned gOff = chunk * 16u;
      asm volatile("global_load_async_to_lds_b128 %0, %1, %2"
                   :: "v"(dOff), "v"(gOff), "s"(Wskip) : "memory");
    }
  }
  asm volatile("s_wait_asynccnt 0x0" ::: "memory");
  __syncthreads();  // all waves see the staged weights

  int wave = blockIdx.x * 8 + (threadIdx.x >> 5);
  if (wave >= nTiles) return;                 // wave-uniform guard
  int lane = threadIdx.x & 31;
  int row  = wave * 16 + (lane & 15);
  int kp   = (lane >> 4) * 2;
  int col  = lane & 15;

  v2f A[16];
  const float* hrow = h + row * 64;
#pragma unroll
  for (int ks = 0; ks < 16; ++ks) {
    A[ks].x = hrow[ks * 4 + kp];
    A[ks].y = hrow[ks * 4 + kp + 1];
  }
  const float* bs[4] = {b0, b1, b2, b3};
  float* os[4] = {o0, o1, o2, o3};
#pragma unroll
  for (int m = 0; m < 4; ++m) {
    const float* Wm = &smW[m * 4096];
    const float* bb = bs[m];
    float* oo = os[m];
#pragma unroll
    for (int ct = 0; ct < 4; ++ct) {
      int c0 = ct * 16;
      float bv = bb[c0 + col];
      v8f acc = {bv, bv, bv, bv, bv, bv, bv, bv};
#pragma unroll
      for (int ks = 0; ks < 16; ++ks) {
        v2f B;
        B.x = Wm[(ks * 4 + kp) * 64 + c0 + col];
        B.y = Wm[(ks * 4 + kp + 1) * 64 + c0 + col];
        acc = wmma4(A[ks], B, acc);
      }
#pragma unroll
      for (int r = 0; r < 8; ++r) {
        int rr = wave * 16 + (lane >> 4) * 8 + r;
        oo[rr * 64 + c0 + col] = acc[r];
      }
    }
  }
}

// ---------------------------------------------------------------------------
// fills
// ---------------------------------------------------------------------------
__global__ void k_fill_f32(float* p, float v, int n) {
  int i = blockIdx.x * blockDim.x + threadIdx.x;
  if (i < n) p[i] = v;
}
__global__ void k_fill_u32(unsigned* p, unsigned v, int n) {
  int i = blockIdx.x * blockDim.x + threadIdx.x;
  if (i < n) p[i] = v;
}

// ---------------------------------------------------------------------------
// Edge pass 1: alpha[e,h] = scale * <q[dst], k[src]+ea*we>, atomic segment max
// ---------------------------------------------------------------------------
__global__ void k_edge_alpha(const float* __restrict__ q,
                             const float* __restrict__ k,
                             const float* __restrict__ ea,
                             const float* __restrict__ we,
                             const int* __restrict__ src,
                             const int* __restrict__ dst,
                             float* __restrict__ alphaBuf,
                             unsigned* __restrict__ amax, int nEH) {
  int t = blockIdx.x * blockDim.x + threadIdx.x;
  if (t >= nEH) return;
  int e = t >> 2, hh = t & 3;
  int s = src[e], d = dst[e];
  float a = ea[e];
  const float* qp = q + d * 64 + hh * 16;
  const float* kp = k + s * 64 + hh * 16;
  const float* wp = we + hh * 16;
  float acc = 0.f;
#pragma unroll
  for (int c = 0; c < 16; ++c) acc += qp[c] * (kp[c] + a * wp[c]);
  acc *= 0.25f;  // 1/sqrt(16)
  alphaBuf[t] = acc;
  atomicMax(&amax[d * 4 + hh], encf(acc));
}

// Edge pass 2: ex = exp(alpha - amax[dst]); segment sum into den
__global__ void k_edge_exp(const int* __restrict__ dst,
                           float* __restrict__ alphaBuf,
                           const unsigned* __restrict__ amax,
                           float* __restrict__ den, int nEH) {
  int t = blockIdx.x * blockDim.x + threadIdx.x;
  if (t >= nEH) return;
  int e = t >> 2, hh = t & 3;
  int d = dst[e];
  float ex = expf(alphaBuf[t] - decf(amax[d * 4 + hh]));
  alphaBuf[t] = ex;  // reuse buffer (same-thread in-place)
  atomicAdd(&den[d * 4 + hh], ex);
}

// Edge pass 3: agg[dst] += (ex/(den+1e-16)) * (v[src] + ea*we), 4 chans/thread
__global__ void k_edge_msg(const float* __restrict__ v,
                           const float* __restrict__ ea,
                           const float* __restrict__ we,
                           const int* __restrict__ src,
                           const int* __restrict__ dst,
                           const float* __restrict__ exbuf,
                           const float* __restrict__ den,
                           float* __restrict__ agg, int nTot) {
  int t = blockIdx.x * blockDim.x + threadIdx.x;
  if (t >= nTot) return;
  int e = t >> 4, cg = t & 15;
  int c0 = cg * 4, hh = c0 >> 4;
  int s = src[e], d = dst[e];
  float w = exbuf[e * 4 + hh] / (den[d * 4 + hh] + 1e-16f);
  float a = ea[e];
#pragma unroll
  for (int i = 0; i < 4; ++i)
    atomicAdd(&agg[d * 64 + c0 + i], w * (v[s * 64 + c0 + i] + a * we[c0 + i]));
}

// ---------------------------------------------------------------------------
// Node update: gated skip + gelu + residual + LayerNorm (in-place on h)
// ---------------------------------------------------------------------------
__global__ void k_node_update(float* __restrict__ h,
                              const float* __restrict__ agg,
                              const float* __restrict__ xr,
                              const float* __restrict__ Wb,
                              const float* __restrict__ lng,
                              const float* __restrict__ lnb, int n) {
  int i = blockIdx.x * blockDim.x + threadIdx.x;
  if (i >= n) return;
  float z = 0.f;
#pragma unroll
  for (int c = 0; c < 64; ++c) {
    float o = agg[i * 64 + c], r = xr[i * 64 + c];
    z += o * Wb[c] + r * Wb[64 + c] + (o - r) * Wb[128 + c];
  }
  float beta = 1.0f / (1.0f + expf(-z));
  float tmp[64];
  float mu = 0.f, sq = 0.f;
#pragma unroll
  for (int c = 0; c < 64; ++c) {
    float o = agg[i * 64 + c], r = xr[i * 64 + c];
    float u = beta * r + (1.0f - beta) * o;
    u = gelu_f(u) + h[i * 64 + c];
    tmp[c] = u;
    mu += u;
    sq += u * u;
  }
  mu *= (1.0f / 64.0f);
  float var = sq * (1.0f / 64.0f) - mu * mu;
  float inv = rsqrtf(var + 1e-5f);
#pragma unroll
  for (int c = 0; c < 64; ++c)
    h[i * 64 + c] = (tmp[c] - mu) * inv * lng[c] + lnb[c];
}

// ---------------------------------------------------------------------------
// Per-graph mean/max reduction
// ---------------------------------------------------------------------------
__global__ void k_batch_reduce(const float* __restrict__ h,
                               const int* __restrict__ batch,
                               float* __restrict__ gsum,
                               unsigned* __restrict__ gmax,
                               float* __restrict__ cnt, int nTot) {
  int t = blockIdx.x * blockDim.x + threadIdx.x;
  if (t >= nTot) return;
  int i = t >> 6, c = t & 63;
  int g = batch[i];
  float v = h[i * 64 + c];
  atomicAdd(&gsum[g * 64 + c], v);
  atomicMax(&gmax[g * 64 + c], encf(v));
  if (c == 0) atomicAdd(&cnt[g], 1.0f);
}

// Readout MLP: [128]->64 gelu ->32 gelu ->1 ; G=128 threads total
__global__ void k_readout(const float* __restrict__ gsum,
                          const unsigned* __restrict__ gmax,
                          const float* __restrict__ cnt,
                          const float* __restrict__ r1W,
                          const float* __restrict__ r1b,
                          const float* __restrict__ r2W,
                          const float* __restrict__ r2b,
                          const float* __restrict__ r3W,
                          const float* __restrict__ r3b,
                          float* __restrict__ out, int G) {
  int g = blockIdx.x * blockDim.x + threadIdx.x;
  if (g >= G) return;
  float rep[128];
  float c = fmaxf(cnt[g], 1.0f);
  for (int j = 0; j < 64; ++j) rep[j] = gsum[g * 64 + j] / c;
  for (int j = 0; j < 64; ++j) rep[64 + j] = decf(gmax[g * 64 + j]);
  float h1[64];
  for (int o = 0; o < 64; ++o) {
    float z = r1b[o];
    for (int j = 0; j < 128; ++j) z += rep[j] * r1W[j * 64 + o];
    h1[o] = gelu_f(z);
  }
  float h2[32];
  for (int o = 0; o < 32; ++o) {
    float z = r2b[o];
    for (int j = 0; j < 64; ++j) z += h1[j] * r2W[j * 32 + o];
    h2[o] = gelu_f(z);
  }
  float z = r3b[0];
  for (int j = 0; j < 32; ++j) z += h2[j] * r3W[j];
  out[g] = z;
}

// ---------------------------------------------------------------------------
extern "C" void kernel_launch(void* const* d_in, const int* in_sizes, int n_in,
                              void* d_out, int out_size, void* d_ws,
                              size_t ws_size, hipStream_t stream) {
  (void)in_sizes; (void)n_in; (void)out_size; (void)ws_size;
  const float* x       = (const float*)d_in[0];
  const int*   eidx    = (const int*)d_in[1];
  const float* ea      = (const float*)d_in[2];
  const int*   batch   = (const int*)d_in[3];
  const float* in_W    = (const float*)d_in[4];
  const float* in_b    = (const float*)d_in[5];
  const float* in_ln_g = (const float*)d_in[6];
  const float* in_ln_b = (const float*)d_in[7];
  const float* W_qkv   = (const float*)d_in[8];
  const float* b_qkv   = (const float*)d_in[9];
  const float* W_edge  = (const float*)d_in[10];
  const float* W_skip  = (const float*)d_in[11];
  const float* b_skip  = (const float*)d_in[12];
  const float* W_beta  = (const float*)d_in[13];
  const float* ln_g    = (const float*)d_in[14];
  const float* ln_b    = (const float*)d_in[15];
  const float* r1W     = (const float*)d_in[16];
  const float* r1b     = (const float*)d_in[17];
  const float* r2W     = (const float*)d_in[18];
  const float* r2b     = (const float*)d_in[19];
  const float* r3W     = (const float*)d_in[20];
  const float* r3b     = (const float*)d_in[21];
  float* out = (float*)d_out;

  const int* src = eidx;
  const int* dst = eidx + N_EDGES;

  char* wp = (char*)d_ws;
  auto alloc = [&](size_t bytes) {
    void* p = wp;
    wp += (bytes + 255) & ~(size_t)255;
    return p;
  };
  float*    h     = (float*)alloc((size_t)N_NODES * 64 * 4);
  float*    q     = (float*)alloc((size_t)N_NODES * 64 * 4);
  float*    k     = (float*)alloc((size_t)N_NODES * 64 * 4);
  float*    v     = (float*)alloc((size_t)N_NODES * 64 * 4);
  float*    xr    = (float*)alloc((size_t)N_NODES * 64 * 4);
  float*    agg   = (float*)alloc((size_t)N_NODES * 64 * 4);
  float*    albuf = (float*)alloc((size_t)N_EDGES * 4 * 4);
  float*    den   = (float*)alloc((size_t)N_NODES * 4 * 4);
  unsigned* amax  = (unsigned*)alloc((size_t)N_NODES * 4 * 4);
  float*    gsum  = (float*)alloc((size_t)NGROUPS * 64 * 4);
  unsigned* gmax  = (unsigned*)alloc((size_t)NGROUPS * 64 * 4);
  float*    cnt   = (float*)alloc((size_t)NGROUPS * 4);

  const int tileBlocks = (NODE_TILES + 7) / 8;  // 8 waves/block
  k_inproj<<<tileBlocks, 256, 0, stream>>>(x, in_W, in_b, in_ln_g, in_ln_b, h,
                                           NODE_TILES);

  for (int l = 0; l < 3; ++l) {
    const float* Wq = W_qkv + (size_t)l * 3 * 4096;   // q,k,v contiguous
    const float* Ws = W_skip + (size_t)l * 4096;
    const float* bq = b_qkv + (size_t)l * 3 * 64;
    const float* bk = bq + 64;
    const float* bv = bk + 64;
    const float* bs = b_skip + (size_t)l * 64;
    const float* we = W_edge + (size_t)l * 64;
    const float* Wb = W_beta + (size_t)l * 192;
    const float* lg = ln_g + (size_t)l * 64;
    const float* lb = ln_b + (size_t)l * 64;

    k_gemm4<<<tileBlocks, 256, 0, stream>>>(h, Wq, Ws, bq, bk, bv, bs, q, k, v,
                                            xr, NODE_TILES);

    k_fill_u32<<<(N_NODES * 4 + 255) / 256, 256, 0, stream>>>(amax, ENC_NEG_INF,
                                                              N_NODES * 4);
    k_fill_f32<<<(N_NODES * 4 + 255) / 256, 256, 0, stream>>>(den, 0.f,
                                                              N_NODES * 4);
    k_fill_f32<<<(N_NODES * 64 + 255) / 256, 256, 0, stream>>>(agg, 0.f,
                                                               N_NODES * 64);

    int nEH = N_EDGES * 4;
    k_edge_alpha<<<(nEH + 255) / 256, 256, 0, stream>>>(q, k, ea, we, src, dst,
                                                        albuf, amax, nEH);
    k_edge_exp<<<(nEH + 255) / 256, 256, 0, stream>>>(dst, albuf, amax, den,
                                                      nEH);
    int nEC = N_EDGES * 16;
    k_edge_msg<<<(nEC + 255) / 256, 256, 0, stream>>>(v, ea, we, src, dst,
                                                      albuf, den, agg, nEC);

    k_node_update<<<(N_NODES + 127) / 128, 128, 0, stream>>>(h, agg, xr, Wb, lg,
                                                             lb, N_NODES);
  }

  k_fill_f32<<<(NGROUPS * 64 + 255) / 256, 256, 0, stream>>>(gsum, 0.f,
                                                             NGROUPS * 64);
  k_fill_u32<<<(NGROUPS * 64 + 255) / 256, 256, 0, stream>>>(gmax, ENC_NEG_INF,
                                                             NGROUPS * 64);
  k_fill_f32<<<1, 256, 0, stream>>>(cnt, 0.f, NGROUPS);
  k_batch_reduce<<<(N_NODES * 64 + 255) / 256, 256, 0, stream>>>(
      h, batch, gsum, gmax, cnt, N_NODES * 64);
  k_readout<<<1, 128, 0, stream>>>(gsum, gmax, cnt, r1W, r1b, r2W, r2b, r3W,
                                   r3b, out, NGROUPS);
}